// ScaledDotProductAttention_Operator_59047210385410
// MI455X (gfx1250) — compile-verified
//
#include <hip/hip_runtime.h>

typedef __attribute__((ext_vector_type(2))) float v2f;
typedef __attribute__((ext_vector_type(8))) float v8f;

#define BN   16      // B*nH
#define TT   64      // T tokens
#define FF   8192    // H*W*C features
#define SLD  68      // padded LDS row stride (even, 4r-bank spread -> conflict free b64)

// ---------------------------------------------------------------------------
// Kernel 1: S[bn][p][q] = (1/1024) * sum_f Q[bn][p][f] * K[bn][q][f]
// One block per head. 8 waves; wave w computes tiles (ti_a, tj) and (ti_a+2, tj)
// of the 4x4 tile grid, sharing the B operand (K rows tj*16..tj*16+15).
// WMMA f32 16x16x4 operand layout (ISA 7.12.2): lane L holds row (L&15),
// K-offset 2*(L>>4); VGPR0/1 = consecutive K pair -> one float2 per operand.
// ---------------------------------------------------------------------------
__global__ void __launch_bounds__(256)
qk_dot_kernel(const float* __restrict__ Q,
              const float* __restrict__ K,
              float* __restrict__ S) {
  const int bn   = blockIdx.x;
  const int lane = threadIdx.x & 31;
  const int wave = threadIdx.x >> 5;
  const int row  = lane & 15;
  const int koff = (lane >> 4) << 1;

  const float* q = Q + (size_t)bn * TT * FF;
  const float* k = K + (size_t)bn * TT * FF;

  const int tj   = wave & 3;
  const int ti_a = wave >> 2;      // 0 or 1
  const int ti_b = ti_a + 2;       // 2 or 3

  const float* qa = q + (size_t)(ti_a * 16 + row) * FF + koff;
  const float* qb = q + (size_t)(ti_b * 16 + row) * FF + koff;
  const float* kb = k + (size_t)(tj   * 16 + row) * FF + koff;

  v8f acc_a = {};
  v8f acc_b = {};
#pragma unroll 4
  for (int kk = 0; kk < FF; kk += 4) {
    v2f a0 = *(const v2f*)(qa + kk);
    v2f a1 = *(const v2f*)(qb + kk);
    v2f b  = *(const v2f*)(kb + kk);
    acc_a = __builtin_amdgcn_wmma_f32_16x16x4_f32(false, a0, false, b,
                                                  (short)0, acc_a, false, false);
    acc_b = __builtin_amdgcn_wmma_f32_16x16x4_f32(false, a1, false, b,
                                                  (short)0, acc_b, false, false);
  }

  // C/D layout: VGPR r -> M = r (lanes 0-15) or M = 8+r (lanes 16-31); N = lane&15.
  const float scale = 1.0f / 1024.0f;   // /= H*W
  float* sb = S + (size_t)bn * TT * TT;
  const int qcol = tj * 16 + row;
  const int half = (lane >> 4) * 8;
#pragma unroll
  for (int r = 0; r < 8; ++r) {
    sb[(ti_a * 16 + half + r) * TT + qcol] = acc_a[r] * scale;
    sb[(ti_b * 16 + half + r) * TT + qcol] = acc_b[r] * scale;
  }
}

// ---------------------------------------------------------------------------
// Kernel 2: O[bn][p][f] = gauss[f>>3] * sum_q S[bn][p][q] * V[bn][q][f]
// Grid: BN * 64 blocks; each of the 8 waves owns one 16-wide f-tile and
// computes all four 16-row p-blocks (M=64, N=16, K=64 -> 64 WMMAs / wave).
// S is staged in padded LDS for conflict-free A-operand reads.
// ---------------------------------------------------------------------------
__global__ void __launch_bounds__(256)
sv_gauss_kernel(const float* __restrict__ S,
                const float* __restrict__ V,
                float* __restrict__ O) {
  __shared__ float Sl[TT * SLD];

  const int bn    = blockIdx.x >> 6;
  const int chunk = blockIdx.x & 63;
  const int tid   = threadIdx.x;
  const int lane  = tid & 31;
  const int wave  = tid >> 5;

  const float* sg = S + (size_t)bn * TT * TT;
  for (int i = tid; i < TT * TT; i += 256) {
    Sl[(i >> 6) * SLD + (i & 63)] = sg[i];
  }
  __syncthreads();

  const int ntile = chunk * 8 + wave;   // 0..511
  const int f0    = ntile * 16;
  const int row   = lane & 15;
  const int koff  = (lane >> 4) << 1;

  const float* v = V + (size_t)bn * TT * FF;

  v8f acc[4] = {};
  for (int kk = 0; kk < TT; kk += 4) {
    v2f b;
    b.x = v[(size_t)(kk + koff)     * FF + f0 + row];
    b.y = v[(size_t)(kk + koff + 1) * FF + f0 + row];
#pragma unroll
    for (int pb = 0; pb < 4; ++pb) {
      v2f a = *(const v2f*)(&Sl[(pb * 16 + row) * SLD + kk + koff]);
      acc[pb] = __builtin_amdgcn_wmma_f32_16x16x4_f32(false, a, false, b,
                                                      (short)0, acc[pb], false, false);
    }
  }

  // Per-lane Gaussian factor: column f = f0 + row, pixel m = f >> 3.
  const int f = f0 + row;
  const int m = f >> 3;
  const float dw = (float)(m & 31) - 15.5f;
  const float dh = (float)(m >> 5) - 15.5f;
  float sx = 0.0f;
#pragma unroll
  for (int x = 0; x < 32; ++x) {
    float d = (float)x - 15.5f;
    sx += __expf(-d * d * (1.0f / 50.0f));   // sigma=5 -> 1/(2*sigma^2) = 1/50
  }
  const float g = __expf(-(dw * dw + dh * dh) * (1.0f / 50.0f)) / (sx * sx);

  float* o = O + (size_t)bn * TT * FF;
  const int half = (lane >> 4) * 8;
#pragma unroll
  for (int pb = 0; pb < 4; ++pb) {
#pragma unroll
    for (int r = 0; r < 8; ++r) {
      const int p = pb * 16 + half + r;
      o[(size_t)p * FF + f] = acc[pb][r] * g;
    }
  }
}

extern "C" void kernel_launch(void* const* d_in, const int* in_sizes, int n_in,
                              void* d_out, int out_size, void* d_ws, size_t ws_size,
                              hipStream_t stream) {
  const float* Q = (const float*)d_in[0];
  const float* K = (const float*)d_in[1];
  const float* V = (const float*)d_in[2];
  float* O = (float*)d_out;
  float* S = (float*)d_ws;   // needs 16*64*64*4 = 256 KB scratch

  qk_dot_kernel<<<BN, 256, 0, stream>>>(Q, K, S);
  sv_gauss_kernel<<<BN * 64, 256, 0, stream>>>(S, V, O);
}